// BatchelorGPUNUFFTAdj_80814104642454
// MI455X (gfx1250) — compile-verified
//
#include <hip/hip_runtime.h>
#include <math.h>

// Problem constants (from reference)
#define NXc 320
#define NCc 8
#define NTc 8
#define Mc  65536
#define Gc  640      // OS * NX
#define Sc  160      // (G - NX) / 2

typedef __attribute__((ext_vector_type(2))) float v2f;
typedef __attribute__((ext_vector_type(8))) float v8f;
typedef unsigned int __attribute__((ext_vector_type(4))) u32x4;
typedef int  __attribute__((ext_vector_type(4))) i32x4;
typedef int  __attribute__((ext_vector_type(8))) i32x8;

#define PI_F 3.14159265358979323846f

#if defined(__has_builtin)
#if __has_builtin(__builtin_amdgcn_tensor_load_to_lds) && __has_builtin(__builtin_amdgcn_s_wait_tensorcnt)
#define HAVE_TDM 1
#endif
#endif

// ---------------------------------------------------------------------------
// Build cropped/shift-folded/deapodized inverse-DFT matrix W (320 x 640),
// W[o,i] = (-1)^(o+i) * exp(+2*pi*j*(160+o)*i/640) / (640 * d[o])
// with d[o] = sinc((o-160)/640)^2. Exact integer phase reduction mod 640.
// ---------------------------------------------------------------------------
__global__ __launch_bounds__(256) void init_w_kernel(float* __restrict__ wre,
                                                     float* __restrict__ wim) {
    int idx = blockIdx.x * 256 + threadIdx.x;
    if (idx >= NXc * Gc) return;
    int o = idx / Gc;
    int i = idx - o * Gc;
    int r = ((Sc + o) * i) % Gc;                 // exact phase in [0,640)
    float theta = (2.0f * PI_F / (float)Gc) * (float)r;
    float x = (float)(o - NXc / 2) / (float)Gc;
    float sc = (x == 0.0f) ? 1.0f : (sinf(PI_F * x) / (PI_F * x));
    float scale = 1.0f / ((float)Gc * sc * sc);  // 1/G and 1/d[o] folded in
    float sign = ((o + i) & 1) ? -1.0f : 1.0f;
    wre[idx] = sign * cosf(theta) * scale;
    wim[idx] = sign * sinf(theta) * scale;
}

// ---------------------------------------------------------------------------
// Density-compensated bilinear gridding (scatter with f32 atomics).
// One thread per sample, loops over the 8 coils. Grid is L2-resident (26MB).
// ---------------------------------------------------------------------------
__global__ __launch_bounds__(256) void grid_kernel(const float* __restrict__ ksr,
                                                   const float* __restrict__ ksi,
                                                   const float* __restrict__ traj,
                                                   const float* __restrict__ dcf,
                                                   float* __restrict__ gre,
                                                   float* __restrict__ gim,
                                                   int t) {
    int m = blockIdx.x * 256 + threadIdx.x;
    if (m >= Mc) return;
    float tx = traj[(m * 2 + 0) * NTc + t];
    float ty = traj[(m * 2 + 1) * NTc + t];
    float w  = dcf[m * NTc + t];
    float u = (tx + 0.5f) * (float)Gc;
    float v = (ty + 0.5f) * (float)Gc;
    float u0 = floorf(u), v0 = floorf(v);
    float du = u - u0, dv = v - v0;
    int i0 = (int)u0; i0 %= Gc; if (i0 < 0) i0 += Gc;
    int j0 = (int)v0; j0 %= Gc; if (j0 < 0) j0 += Gc;
    int i1 = (i0 + 1 == Gc) ? 0 : i0 + 1;
    int j1 = (j0 + 1 == Gc) ? 0 : j0 + 1;
    float w00 = (1.0f - du) * (1.0f - dv) * w;
    float w10 = du * (1.0f - dv) * w;
    float w01 = (1.0f - du) * dv * w;
    float w11 = du * dv * w;
    int o00 = i0 * Gc + j0, o10 = i1 * Gc + j0;
    int o01 = i0 * Gc + j1, o11 = i1 * Gc + j1;
#pragma unroll
    for (int c = 0; c < NCc; ++c) {
        float re = ksr[c * Mc + m];
        float im = ksi[c * Mc + m];
        int base = c * Gc * Gc;
        atomicAdd(&gre[base + o00], re * w00); atomicAdd(&gim[base + o00], im * w00);
        atomicAdd(&gre[base + o10], re * w10); atomicAdd(&gim[base + o10], im * w10);
        atomicAdd(&gre[base + o01], re * w01); atomicAdd(&gim[base + o01], im * w01);
        atomicAdd(&gre[base + o11], re * w11); atomicAdd(&gim[base + o11], im * w11);
    }
}

// ---------------------------------------------------------------------------
// TDM: DMA a 640-row x 16-col f32 strip (row stride = 640 elems) from global
// into LDS, compacted to 16-float rows. D# per CDNA5 ISA:
//   group0: count=1 | lds_addr | global_addr[56:0] | type=2
//   group1: data_size=4B, tensor_dim0=16, tensor_dim1=640,
//           tile_dim0=16, tile_dim1=640, tensor_dim0_stride=640
// Issued from one wave; TENSORcnt waited there, block released via barrier.
// ---------------------------------------------------------------------------
#if defined(HAVE_TDM)
__device__ __forceinline__ void tdm_load_strip16(const float* src, float* ldsDst) {
    unsigned long long ga = (unsigned long long)(size_t)src;
    unsigned int ldsOff = (unsigned int)(size_t)ldsDst;   // LDS aperture: addr[31:0] = LDS byte offset
    u32x4 g0;
    g0.x = 1u;                                            // count=1, is_restore=0, gather off
    g0.y = ldsOff;                                        // lds_addr (bytes)
    g0.z = (unsigned int)(ga & 0xFFFFFFFFull);            // global_addr[31:0]
    g0.w = (unsigned int)((ga >> 32) & 0x01FFFFFFull) | (2u << 30);  // addr[56:32] | type=2
    i32x8 g1;
    g1[0] = (2 << 16);                                    // wg_mask=0, data_size=2 (4 bytes)
    g1[1] = (16 << 16);                                   // tensor_dim0 = 16 (bits 79:48, low part)
    g1[2] = (640 << 16);                                  // tensor_dim0 hi=0 | tensor_dim1=640 (bits 95:80)
    g1[3] = (16 << 16);                                   // tensor_dim1 hi=0 | tile_dim0=16 (bits 127:112)
    g1[4] = 640;                                          // tile_dim1=640, tile_dim2=0
    g1[5] = 640;                                          // tensor_dim0_stride = 640 (48-bit, low word)
    g1[6] = 0;                                            // stride hi | tensor_dim1_stride lo
    g1[7] = 0;
    i32x4 zz4 = {0, 0, 0, 0};
#if defined(__clang_major__) && (__clang_major__ >= 23)
    // upstream clang-23 / therock-10.0 lane: 6-arg form (extra i32x8 group)
    i32x8 zz8 = {0, 0, 0, 0, 0, 0, 0, 0};
    __builtin_amdgcn_tensor_load_to_lds(g0, g1, zz4, zz4, zz8, 0);
#else
    // ROCm 7.2 / clang-22 lane: 5-arg form
    __builtin_amdgcn_tensor_load_to_lds(g0, g1, zz4, zz4, 0);
#endif
}
#endif

// ---------------------------------------------------------------------------
// GEMM1: T[c,o,l] = sum_k W[o,k] * g[c,k,l]   (complex, M=320 N=640 K=640)
// One block per (coil, 16-col strip): B strip staged once (TDM async DMA ->
// LDS, 80KB), 8 waves sweep the 20 row tiles with V_WMMA_F32_16X16X4_F32.
// 4 accumulators implement the complex product (F32 WMMA has no A/B negate).
// ---------------------------------------------------------------------------
__global__ __launch_bounds__(256) void gemm1_kernel(const float* __restrict__ wre,
                                                    const float* __restrict__ wim,
                                                    const float* __restrict__ gre,
                                                    const float* __restrict__ gim,
                                                    float* __restrict__ tre,
                                                    float* __restrict__ tim) {
    const int OT = NXc / 16;   // 20 row tiles
    const int LT = Gc / 16;    // 40 col strips
    __shared__ float ldsRe[Gc * 16];   // 40KB
    __shared__ float ldsIm[Gc * 16];   // 40KB
    int wave = threadIdx.x >> 5;
    int lane = threadIdx.x & 31;
    int c  = blockIdx.x / LT;          // 8 * 40 = 320 blocks
    int lt = blockIdx.x - c * LT;
    const float* greC = gre + c * (Gc * Gc) + lt * 16;
    const float* gimC = gim + c * (Gc * Gc) + lt * 16;

#if defined(HAVE_TDM)
    if (wave == 0) {
        tdm_load_strip16(greC, ldsRe);
        tdm_load_strip16(gimC, ldsIm);
        __builtin_amdgcn_s_wait_tensorcnt(0);
    }
#else
    for (int idx = threadIdx.x; idx < Gc * 16; idx += 256) {
        int r = idx >> 4, col = idx & 15;
        ldsRe[idx] = greC[r * Gc + col];
        ldsIm[idx] = gimC[r * Gc + col];
    }
#endif
    __syncthreads();

    int half = lane >> 4;              // K-pair selector
    int lr   = lane & 15;              // row (A) / col (B)
    int bcol = lt * 16 + lr;

    for (int ot = wave; ot < OT; ot += 8) {     // wave-uniform, EXEC stays full
        int arow = ot * 16 + lr;
        const float* wreRow = wre + arow * Gc;
        const float* wimRow = wim + arow * Gc;
        v8f P = {}, Q = {}, R = {}, Sa = {};
        for (int k = 0; k < Gc; k += 4) {
            int ka = k + half * 2;
            v2f are = *(const v2f*)(wreRow + ka);
            v2f aim = *(const v2f*)(wimRow + ka);
            v2f bre, bim;
            bre.x = ldsRe[ka * 16 + lr];
            bre.y = ldsRe[(ka + 1) * 16 + lr];
            bim.x = ldsIm[ka * 16 + lr];
            bim.y = ldsIm[(ka + 1) * 16 + lr];
            P  = __builtin_amdgcn_wmma_f32_16x16x4_f32(false, are, false, bre, (short)0, P,  false, false);
            Q  = __builtin_amdgcn_wmma_f32_16x16x4_f32(false, aim, false, bim, (short)0, Q,  false, false);
            R  = __builtin_amdgcn_wmma_f32_16x16x4_f32(false, are, false, bim, (short)0, R,  false, false);
            Sa = __builtin_amdgcn_wmma_f32_16x16x4_f32(false, aim, false, bre, (short)0, Sa, false, false);
        }
        v8f Cr = P - Q;
        v8f Ci = R + Sa;
        int obase = c * (NXc * Gc);
#pragma unroll
        for (int r = 0; r < 8; ++r) {
            int m = r + half * 8;
            int off = obase + (ot * 16 + m) * Gc + bcol;
            tre[off] = Cr[r];
            tim[off] = Ci[r];
        }
    }
}

// ---------------------------------------------------------------------------
// GEMM2: im[c,o,q] = sum_l T[c,o,l] * W[q,l]  (complex, M=320 N=320 K=640)
// Both fragments are contiguous b64 loads (T rows, W rows) from L2.
// ---------------------------------------------------------------------------
__global__ __launch_bounds__(256) void gemm2_kernel(const float* __restrict__ wre,
                                                    const float* __restrict__ wim,
                                                    const float* __restrict__ tre,
                                                    const float* __restrict__ tim,
                                                    float* __restrict__ imre,
                                                    float* __restrict__ imim) {
    const int OT = NXc / 16;   // 20
    int wave = threadIdx.x >> 5;
    int lane = threadIdx.x & 31;
    int tile = blockIdx.x * 8 + wave;          // 8 * 20 * 20 = 3200 tiles
    int c  = tile / (OT * OT);
    int r2 = tile - c * (OT * OT);
    int ot = r2 / OT;
    int qt = r2 - ot * OT;
    int half = lane >> 4;
    int lr   = lane & 15;
    int arow = ot * 16 + lr;
    int qcol = qt * 16 + lr;
    const float* treRow = tre + c * (NXc * Gc) + arow * Gc;
    const float* timRow = tim + c * (NXc * Gc) + arow * Gc;
    const float* wreRow = wre + qcol * Gc;
    const float* wimRow = wim + qcol * Gc;
    v8f P = {}, Q = {}, R = {}, Sa = {};
    for (int k = 0; k < Gc; k += 4) {
        int ka = k + half * 2;
        v2f are = *(const v2f*)(treRow + ka);
        v2f aim = *(const v2f*)(timRow + ka);
        v2f bre = *(const v2f*)(wreRow + ka);
        v2f bim = *(const v2f*)(wimRow + ka);
        P  = __builtin_amdgcn_wmma_f32_16x16x4_f32(false, are, false, bre, (short)0, P,  false, false);
        Q  = __builtin_amdgcn_wmma_f32_16x16x4_f32(false, aim, false, bim, (short)0, Q,  false, false);
        R  = __builtin_amdgcn_wmma_f32_16x16x4_f32(false, are, false, bim, (short)0, R,  false, false);
        Sa = __builtin_amdgcn_wmma_f32_16x16x4_f32(false, aim, false, bre, (short)0, Sa, false, false);
    }
    v8f Cr = P - Q;
    v8f Ci = R + Sa;
    int obase = c * (NXc * NXc);
#pragma unroll
    for (int r = 0; r < 8; ++r) {
        int m = r + half * 8;
        int off = obase + (ot * 16 + m) * NXc + qcol;
        imre[off] = Cr[r];
        imim[off] = Ci[r];
    }
}

// ---------------------------------------------------------------------------
// Conjugate coil combine: comb = sum_c conj(csm[c]) * im[c]
// ---------------------------------------------------------------------------
__global__ __launch_bounds__(256) void combine_kernel(const float* __restrict__ csr,
                                                      const float* __restrict__ csi,
                                                      const float* __restrict__ imre,
                                                      const float* __restrict__ imim,
                                                      float* __restrict__ combre,
                                                      float* __restrict__ combim) {
    int idx = blockIdx.x * 256 + threadIdx.x;
    if (idx >= NXc * NXc) return;
    float ar = 0.0f, ai = 0.0f;
#pragma unroll
    for (int c = 0; c < NCc; ++c) {
        float cr = csr[c * (NXc * NXc) + idx];
        float ci = csi[c * (NXc * NXc) + idx];
        float ir = imre[c * (NXc * NXc) + idx];
        float ii = imim[c * (NXc * NXc) + idx];
        ar += cr * ir + ci * ii;       // conj(csm) * im
        ai += cr * ii - ci * ir;
    }
    combre[idx] = ar;
    combim[idx] = ai;
}

// ---------------------------------------------------------------------------
// Bilinear motion warp of the combined frame, accumulated into d_out.
// out layout: (NX, NX, 2) -> [x][y][re/im]
// ---------------------------------------------------------------------------
__global__ __launch_bounds__(256) void warp_acc_kernel(const float* __restrict__ combre,
                                                       const float* __restrict__ combim,
                                                       const float* __restrict__ motions,
                                                       float* __restrict__ out,
                                                       int t) {
    int idx = blockIdx.x * 256 + threadIdx.x;
    if (idx >= NXc * NXc) return;
    int x = idx / NXc;
    int y = idx - x * NXc;
    float f0 = motions[((x * NXc + y) * 2 + 0) * NTc + t];
    float f1 = motions[((x * NXc + y) * 2 + 1) * NTc + t];
    float xs = (float)x + f0;
    float ys = (float)y + f1;
    xs = fminf(fmaxf(xs, 0.0f), (float)(NXc - 1));
    ys = fminf(fmaxf(ys, 0.0f), (float)(NXc - 1));
    int x0 = (int)floorf(xs);
    int y0 = (int)floorf(ys);
    int x1 = (x0 + 1 < NXc) ? x0 + 1 : NXc - 1;
    int y1 = (y0 + 1 < NXc) ? y0 + 1 : NXc - 1;
    float dx = xs - (float)x0;
    float dy = ys - (float)y0;
    float w00 = (1.0f - dx) * (1.0f - dy);
    float w10 = dx * (1.0f - dy);
    float w01 = (1.0f - dx) * dy;
    float w11 = dx * dy;
    int o00 = x0 * NXc + y0, o10 = x1 * NXc + y0;
    int o01 = x0 * NXc + y1, o11 = x1 * NXc + y1;
    float sr = w00 * combre[o00] + w10 * combre[o10] + w01 * combre[o01] + w11 * combre[o11];
    float si = w00 * combim[o00] + w10 * combim[o10] + w01 * combim[o01] + w11 * combim[o11];
    out[idx * 2 + 0] += sr;
    out[idx * 2 + 1] += si;
}

// ---------------------------------------------------------------------------
// Host-side orchestration (all on `stream`, graph-capture safe).
// ---------------------------------------------------------------------------
extern "C" void kernel_launch(void* const* d_in, const int* in_sizes, int n_in,
                              void* d_out, int out_size, void* d_ws, size_t ws_size,
                              hipStream_t stream) {
    const float* ksr     = (const float*)d_in[0];  // (NC, M)
    const float* ksi     = (const float*)d_in[1];  // (NC, M)
    const float* traj    = (const float*)d_in[2];  // (M, 2, NT)
    const float* csr     = (const float*)d_in[3];  // (NC, NX, NX)
    const float* csi     = (const float*)d_in[4];  // (NC, NX, NX)
    const float* dcf     = (const float*)d_in[5];  // (M, NT)
    const float* motions = (const float*)d_in[6];  // (NX, NX, 2, NT)
    float* out = (float*)d_out;                    // (NX, NX, 2)

    // Workspace layout (floats)
    float* ws = (float*)d_ws;
    size_t off = 0;
    float* wre    = ws + off; off += (size_t)NXc * Gc;        // 204800
    float* wim    = ws + off; off += (size_t)NXc * Gc;
    float* gre    = ws + off; off += (size_t)NCc * Gc * Gc;   // 3276800
    float* gim    = ws + off; off += (size_t)NCc * Gc * Gc;
    float* tre    = ws + off; off += (size_t)NCc * NXc * Gc;  // 1638400
    float* tim    = ws + off; off += (size_t)NCc * NXc * Gc;
    float* imre   = ws + off; off += (size_t)NCc * NXc * NXc; // 819200
    float* imim   = ws + off; off += (size_t)NCc * NXc * NXc;
    float* combre = ws + off; off += (size_t)NXc * NXc;
    float* combim = ws + off; off += (size_t)NXc * NXc;

    // Build DFT/deapodization matrix once per call
    init_w_kernel<<<(NXc * Gc + 255) / 256, 256, 0, stream>>>(wre, wim);

    // Zero the frame accumulator (harness poisons d_out)
    (void)hipMemsetAsync(out, 0, (size_t)NXc * NXc * 2 * sizeof(float), stream);

    for (int t = 0; t < NTc; ++t) {
        (void)hipMemsetAsync(gre, 0, (size_t)NCc * Gc * Gc * sizeof(float), stream);
        (void)hipMemsetAsync(gim, 0, (size_t)NCc * Gc * Gc * sizeof(float), stream);

        grid_kernel<<<Mc / 256, 256, 0, stream>>>(ksr, ksi, traj, dcf, gre, gim, t);

        // One block per (coil, 16-col strip): 8 * 40 = 320 blocks
        gemm1_kernel<<<NCc * (Gc / 16), 256, 0, stream>>>(wre, wim, gre, gim, tre, tim);
        // 8 coils * 20 * 20 tiles = 3200 wave-tiles, 8 waves/block
        gemm2_kernel<<<3200 / 8, 256, 0, stream>>>(wre, wim, tre, tim, imre, imim);

        combine_kernel<<<(NXc * NXc) / 256, 256, 0, stream>>>(csr, csi, imre, imim, combre, combim);
        warp_acc_kernel<<<(NXc * NXc) / 256, 256, 0, stream>>>(combre, combim, motions, out, t);
    }
}